// RoiPooling_56307021250934
// MI455X (gfx1250) — compile-verified
//
#include <hip/hip_runtime.h>
#include <stdint.h>

// ROI bilinear pooling for MI455X (gfx1250).
//
// Memory-bound: ~300 MB total traffic, ~120 MFLOP. The 164 MB feature map fits
// in the 192 MB L2, so corner gathers re-hit L2 across the 14700 output cells.
// No matmul structure -> WMMA inapplicable; the CDNA5 path used is the async
// global->LDS streaming engine (GLOBAL_LOAD_ASYNC_TO_LDS_B128 / ASYNCcnt).
// Output stores are non-temporal (TH_STORE_NT) so the 60 MB write-once stream
// does not evict the L2-resident feature map.

#define POOL 7
#define THREADS 256   // 8 wave32s; 256 lanes x float4 = 1024 channels per shot

// native clang vector: required by __builtin_nontemporal_store, same 16B layout
typedef float f32x4 __attribute__((ext_vector_type(4)));

__global__ __launch_bounds__(THREADS) void roi_pool_bilinear_gfx1250(
    const float* __restrict__ feat,   // (H, W, C) f32, C contiguous
    const int*   __restrict__ rois,   // (N, 4) int32: x0, y0, w, h
    float*       __restrict__ out,    // (N, POOL, POOL, C) f32
    int H, int W, int C)
{
    const int cell = blockIdx.x;                 // roi*49 + py*7 + px
    const int roi  = cell / (POOL * POOL);
    const int rem  = cell - roi * POOL * POOL;
    const int py   = rem / POOL;
    const int px   = rem - py * POOL;

    // ---- block-uniform ROI decode (scalar path) ----
    const int4 r = ((const int4*)rois)[roi];
    const int x0 = r.x, y0 = r.y, w = r.z, h = r.w;

    // exact reference arithmetic: ry = py * (h / 7.0f), rx = px * (w / 7.0f)
    const float ry  = (float)py * ((float)h / (float)POOL);
    const float rx  = (float)px * ((float)w / (float)POOL);
    const int   ylo = (int)floorf(ry);
    const int   xlo = (int)floorf(rx);
    const float fy  = ry - (float)ylo;
    const float fx  = rx - (float)xlo;
    const int   yhi = min(ylo + 1, h - 1);
    const int   xhi = min(xlo + 1, w - 1);
    const int   iy0 = min(max(y0 + ylo, 0), H - 1);
    const int   iy1 = min(max(y0 + yhi, 0), H - 1);
    const int   ix0 = min(max(x0 + xlo, 0), W - 1);
    const int   ix1 = min(max(x0 + xhi, 0), W - 1);

    // byte offsets of the 4 corner channel-rows (<= 164 MB, fits u32)
    const uint32_t rowBytes = (uint32_t)C * 4u;
    const uint32_t b00 = (uint32_t)(iy0 * W + ix0) * rowBytes;
    const uint32_t b01 = (uint32_t)(iy0 * W + ix1) * rowBytes;
    const uint32_t b10 = (uint32_t)(iy1 * W + ix0) * rowBytes;
    const uint32_t b11 = (uint32_t)(iy1 * W + ix1) * rowBytes;

    // 4 corners x 256 lanes x 16 B = 16 KB of LDS (WGP has 320 KB)
    __shared__ float smem[4 * THREADS * 4];

    const uint32_t tid    = threadIdx.x;
    const uint32_t lane16 = tid * 16u;            // this lane's 16 B channel chunk
    const bool     active = (lane16 < rowBytes);  // always true for C >= 1024

    // LDS byte address for DS/async ops: low 32 bits of the generic pointer
    // (generic LDS pointer = {aperture_hi, lds_offset}).
    const uint32_t ldsBase = (uint32_t)(uintptr_t)(&smem[0]);
    const uint32_t l0 = ldsBase + 0u * (THREADS * 16u) + lane16;
    const uint32_t l1 = ldsBase + 1u * (THREADS * 16u) + lane16;
    const uint32_t l2 = ldsBase + 2u * (THREADS * 16u) + lane16;
    const uint32_t l3 = ldsBase + 3u * (THREADS * 16u) + lane16;

    if (active) {
        // GVS mode: saddr = 64-bit base (SGPR pair), vaddr = 32-bit per-lane
        // byte offset, vdst = per-lane LDS byte address. Tracked by ASYNCcnt.
        asm volatile("global_load_async_to_lds_b128 %0, %1, %2"
                     :: "v"(l0), "v"(b00 + lane16), "s"(feat) : "memory");
        asm volatile("global_load_async_to_lds_b128 %0, %1, %2"
                     :: "v"(l1), "v"(b01 + lane16), "s"(feat) : "memory");
        asm volatile("global_load_async_to_lds_b128 %0, %1, %2"
                     :: "v"(l2), "v"(b10 + lane16), "s"(feat) : "memory");
        asm volatile("global_load_async_to_lds_b128 %0, %1, %2"
                     :: "v"(l3), "v"(b11 + lane16), "s"(feat) : "memory");
    }

    // Each lane consumes only the LDS bytes it queued itself -> no barrier,
    // just drain this wave's async queue.
    asm volatile("s_wait_asynccnt 0" ::: "memory");

    if (!active) return;

    const float wx1v = fx, wx0v = 1.0f - fx;
    const float wy1v = fy, wy0v = 1.0f - fy;

    const f32x4* sm4 = (const f32x4*)smem;
    const f32x4 f00 = sm4[0 * THREADS + tid];
    const f32x4 f01 = sm4[1 * THREADS + tid];
    const f32x4 f10 = sm4[2 * THREADS + tid];
    const f32x4 f11 = sm4[3 * THREADS + tid];

    // bilinear blend (vectorized over 4 channels)
    const f32x4 top = f00 * wx0v + f01 * wx1v;
    const f32x4 bot = f10 * wx0v + f11 * wx1v;
    const f32x4 res = top * wy0v + bot * wy1v;

    // out[(roi, py, px, c)] flat = cell*C + 4*tid  -> one b128 store per lane.
    // Non-temporal: write-once stream must not evict the L2-resident feature
    // map (164 MB of 192 MB). Lowers to global_store_b128 ... th:TH_STORE_NT.
    f32x4* dst = &((f32x4*)out)[(size_t)cell * (uint32_t)(C / 4) + tid];
    __builtin_nontemporal_store(res, dst);
}

extern "C" void kernel_launch(void* const* d_in, const int* in_sizes, int n_in,
                              void* d_out, int out_size, void* d_ws, size_t ws_size,
                              hipStream_t stream) {
    (void)n_in; (void)out_size; (void)d_ws; (void)ws_size;
    const float* feat = (const float*)d_in[0];   // (1, 200, 200, 1024) f32
    const int*   rois = (const int*)d_in[1];     // (N, 4) int32
    float*       out  = (float*)d_out;           // (N, 7, 7, 1024) f32

    const int H = 200, W = 200;
    const int N = in_sizes[1] / 4;
    const int C = in_sizes[0] / (H * W);         // 1024

    const int blocks = N * POOL * POOL;          // 14700 cells
    roi_pool_bilinear_gfx1250<<<blocks, THREADS, 0, stream>>>(feat, rois, out, H, W, C);
}